// PadToFullGraphEdgeEncoder_70849780514914
// MI455X (gfx1250) — compile-verified
//
#include <hip/hip_runtime.h>
#include <cstdint>

#ifndef __has_builtin
#define __has_builtin(x) 0
#endif

#define AS1 __attribute__((address_space(1)))
#define AS3 __attribute__((address_space(3)))

// 128-bit payload type the async-LDS builtins expect (b128 = 4 dwords)
typedef int v4i __attribute__((vector_size(16)));
typedef AS1 v4i* gptr128_t;  // global (AS1) pointer
typedef AS3 v4i* lptr128_t;  // LDS (AS3) pointer

// ---------------------------------------------------------------------------
// Kernel 1: fill d_out.
//   [0, 2S)        : dense adjacency indices as floats (rows then cols)
//   [2S, 2S + S*D) : zeros (destination of the scatter-add)
// Zero region is written with gfx1250 async LDS->global stores (ASYNCcnt
// path): one 4 KB zeroed LDS tile per block feeds
// global_store_async_from_lds_b128, 16 B per lane (512 B per wave issue).
// All index math is 32-bit unsigned (max flat index 138M < 2^31) so the
// divisions lower to the short reciprocal sequence, not 64-bit long division.
// ---------------------------------------------------------------------------
__global__ __launch_bounds__(256) void fill_kernel(float* __restrict__ out,
                                                   unsigned S,      // B*n*n
                                                   unsigned n, unsigned nn,
                                                   unsigned total4) // out_size/4
{
    __shared__ __align__(16) float ldsz[256 * 4]; // 4 KB of zeros per block
    const unsigned t = threadIdx.x;
    ((float4*)ldsz)[t] = make_float4(0.f, 0.f, 0.f, 0.f);
    __syncthreads();

    const unsigned S2 = 2u * S;
    unsigned i = blockIdx.x * blockDim.x + t;
    if (i < total4) {
        unsigned f = i << 2; // flat float index of this thread's float4
        if (f < S2) {
            // ---- index region: compute 4 values (S is a multiple of 4, so a
            // float4 never straddles the rows/cols boundary) ----
            float4 v;
            float* pv = (float*)&v;
#pragma unroll
            for (int k = 0; k < 4; ++k) {
                unsigned ff = f + (unsigned)k;
                unsigned val;
                if (ff < S) {                       // rows: m/n
                    val = ff / n;
                } else {                            // cols: (m/n^2)*n + m%n
                    unsigned m = ff - S;
                    val = (m / nn) * n + (m % n);
                }
                pv[k] = (float)val;
            }
            ((float4*)out)[i] = v;
        } else {
            // ---- zero region: async LDS -> global store (ASYNCcnt path) ----
#if __has_builtin(__builtin_amdgcn_global_store_async_from_lds_b128)
            gptr128_t gp = (gptr128_t)(uintptr_t)(out + (size_t)f);
            lptr128_t lp = (lptr128_t)(uint32_t)(uintptr_t)(&ldsz[t * 4]);
            __builtin_amdgcn_global_store_async_from_lds_b128(gp, lp, 0, 0);
#else
            ((float4*)out)[i] = make_float4(0.f, 0.f, 0.f, 0.f);
#endif
        }
    }

#if __has_builtin(__builtin_amdgcn_global_store_async_from_lds_b128)
    // Drain ASYNCcnt before the wave retires (S_ENDPGM's implicit wait-idle
    // would also cover this, but be explicit).
#if __has_builtin(__builtin_amdgcn_s_wait_asynccnt)
    __builtin_amdgcn_s_wait_asynccnt(0);
#else
    asm volatile("s_wait_asynccnt 0" ::: "memory");
#endif
#endif
}

// ---------------------------------------------------------------------------
// Kernel 2: scatter-add edge_attr into the zeroed out_val region.
// One thread per float4 of one edge row: 16 lanes cover one 256 B edge row.
// pos = g*n^2 + (r-g*n)*n + (c-g*n) = r*n + c - g*n with g = r/n.
// Duplicate edges require RMW -> global_atomic_add_f32 (order-independent for
// the dominant 2-way duplicate case; deterministic to tolerance otherwise).
// ---------------------------------------------------------------------------
__global__ __launch_bounds__(256) void scatter_kernel(const int* __restrict__ ei,
                                                      const float* __restrict__ ea,
                                                      float* __restrict__ outv,
                                                      unsigned E, unsigned n,
                                                      unsigned Dq, unsigned D)
{
    unsigned tid = blockIdx.x * blockDim.x + threadIdx.x;
    unsigned total = E * Dq;           // 8.4M < 2^31
    if (tid >= total) return;

    unsigned e = tid / Dq;
    unsigned q = tid - e * Dq;

    unsigned r = (unsigned)ei[e];
    unsigned c = (unsigned)ei[E + e];
    unsigned g = r / n;
    unsigned pos = r * n + c - g * n;  // < S = 2^21, fits easily

    const float4 v = ((const float4*)ea)[(size_t)e * Dq + q];
    float* p = outv + (size_t)pos * D + (size_t)q * 4u;

#if defined(__HIP_DEVICE_COMPILE__)
    unsafeAtomicAdd(p + 0, v.x);   // -> global_atomic_add_f32
    unsafeAtomicAdd(p + 1, v.y);
    unsafeAtomicAdd(p + 2, v.z);
    unsafeAtomicAdd(p + 3, v.w);
#else
    atomicAdd(p + 0, v.x);
    atomicAdd(p + 1, v.y);
    atomicAdd(p + 2, v.z);
    atomicAdd(p + 3, v.w);
#endif
}

// ---------------------------------------------------------------------------
// Host launcher. Inputs (setup_inputs order):
//   d_in[0] edge_index int32 [2,E]   d_in[1] edge_attr f32 [E,D]
//   d_in[2] batch int32 [N]          d_in[3] num_graphs   d_in[4] n_per_graph
// All shape params derived from sizes; batch is redundant (batch[r] == r/n).
// ---------------------------------------------------------------------------
extern "C" void kernel_launch(void* const* d_in, const int* in_sizes, int n_in,
                              void* d_out, int out_size, void* d_ws, size_t ws_size,
                              hipStream_t stream) {
    const int*   ei  = (const int*)d_in[0];
    const float* ea  = (const float*)d_in[1];
    float*       out = (float*)d_out;

    const unsigned E = (unsigned)(in_sizes[0] / 2);           // 524288
    const unsigned D = (unsigned)(in_sizes[1] / (int)E);      // 64
    const unsigned N = (unsigned)in_sizes[2];                 // B*n
    const unsigned S = (unsigned)((long long)out_size / (D + 2)); // B*n*n
    const unsigned n = S / N;                                 // 128
    const unsigned Dq = D / 4;

    // Kernel 1: indices + zero-fill (one float4 per thread, 1:1 mapping)
    const unsigned total4 = (unsigned)(out_size / 4);
    const unsigned tb = 256;
    const unsigned fb = (total4 + tb - 1) / tb;
    fill_kernel<<<fb, tb, 0, stream>>>(out, S, n, n * n, total4);

    // Kernel 2: atomic scatter-add of edge attrs
    const unsigned totalS = E * Dq;
    const unsigned sb = (totalS + tb - 1) / tb;
    scatter_kernel<<<sb, tb, 0, stream>>>(ei, ea, out + (size_t)2 * S, E, n, Dq, D);
}